// Net_82884278879026
// MI455X (gfx1250) — compile-verified
//
#include <hip/hip_runtime.h>
#include <hip/hip_bf16.h>
#include <math.h>

// Problem constants (from reference)
#define NSPIN   197
#define DDIM    32
#define NDIM    (NSPIN * DDIM)      // 6304
#define BATCH   8
#define NPAD    16                  // batch padded to WMMA N=16
#define NPATCH  196
#define PDIM    768
#define CGIT    32
#define KC      32                  // k-chunk per LDS stage
#define NCHUNK  (NDIM / KC)         // 197

typedef float v2f __attribute__((ext_vector_type(2)));
typedef float v8f __attribute__((ext_vector_type(8)));

// ---------------------------------------------------------------- utilities
__global__ void k_zero(float* __restrict__ p, long n) {
    long i = blockIdx.x * (long)blockDim.x + threadIdx.x;
    long stride = (long)gridDim.x * blockDim.x;
    for (; i < n; i += stride) p[i] = 0.0f;
}

__device__ inline float gelu_exact(float v) {
    return 0.5f * v * (1.0f + erff(v * 0.70710678118654752440f));
}

// ------------------------------------------------- patch embed + feedforward
// one wave32 per (batch, patch); lane = output channel (DDIM == 32 == wave)
__global__ void __launch_bounds__(256)
k_embed(const float* __restrict__ x,  const float* __restrict__ wp,
        const float* __restrict__ bp, const float* __restrict__ w1,
        const float* __restrict__ b1, const float* __restrict__ w2,
        const float* __restrict__ b2, float* __restrict__ T) {
    int wave = threadIdx.x >> 5, lane = threadIdx.x & 31;
    int pair = blockIdx.x * 8 + wave;            // 0 .. 1567
    int b = pair / NPATCH, p = pair % NPATCH;
    int ph = p / 14, pw = p % 14;
    const float* xb = x + (long)b * 3 * 224 * 224;
    const float* wrow = wp + lane * PDIM;

    float acc = bp[lane];
    int k = 0;
    for (int p1 = 0; p1 < 16; ++p1) {
        int row = ph * 16 + p1;
        for (int p2 = 0; p2 < 16; ++p2) {
            int colp = pw * 16 + p2;
            #pragma unroll
            for (int c = 0; c < 3; ++c) {
                float xv = xb[((long)c * 224 + row) * 224 + colp]; // broadcast load
                acc += wrow[k] * xv;
                ++k;
            }
        }
    }
    // FF layer 1 (32x32) via cross-lane broadcast of the embed vector
    float h = b1[lane];
    #pragma unroll
    for (int c = 0; c < DDIM; ++c) h += w1[lane * DDIM + c] * __shfl(acc, c, 32);
    h = gelu_exact(h);
    // FF layer 2
    float o = b2[lane];
    #pragma unroll
    for (int c = 0; c < DDIM; ++c) o += w2[lane * DDIM + c] * __shfl(h, c, 32);
    // store into transposed/padded RHS  T[row][batchcol]
    T[((long)(p + 1) * DDIM + lane) * NPAD + b] = o;
}

// -------------------------------------------------- A = I - sym(J_raw)*mask
__global__ void k_buildA(const float* __restrict__ Jr, float* __restrict__ A) {
    const long n = (long)NDIM * NDIM;
    long i = blockIdx.x * (long)blockDim.x + threadIdx.x;
    long stride = (long)gridDim.x * blockDim.x;
    for (; i < n; i += stride) {
        int r  = (int)(i / NDIM), cj = (int)(i % NDIM);
        int si = r >> 5,  c = r & 31;
        int sj = cj >> 5, d = cj & 31;
        float v = 0.0f;
        if (si != sj) {
            float j1 = Jr[(((long)si * NSPIN + sj) * DDIM + c) * DDIM + d];
            float j2 = Jr[(((long)sj * NSPIN + si) * DDIM + d) * DDIM + c];
            v = 0.5f * (j1 + j2);
        }
        A[i] = (r == cj ? 1.0f : 0.0f) - v;   // LAMBDA = 1
    }
}

// ---------------------------------------------------------------- CG init
__global__ void __launch_bounds__(1024)
k_cginit(float* __restrict__ T, float* __restrict__ X, float* __restrict__ R,
         float* __restrict__ Pv, float* __restrict__ rs,
         const float* __restrict__ cls) {
    __shared__ float red[1024];
    int t = threadIdx.x;
    if (t < 256) { int d = t >> 3, b = t & 7; T[d * NPAD + b] = cls[d]; }
    __syncthreads();
    int col = t & 15, g = t >> 4;
    float part = 0.0f;
    for (int r = g; r < NDIM; r += 64) {
        int e = r * NPAD + col;
        float tv = T[e];
        R[e] = tv; Pv[e] = tv; X[e] = 0.0f;
        part += tv * tv;
    }
    red[t] = part; __syncthreads();
    for (int s = 512; s >= 16; s >>= 1) { if (t < s) red[t] += red[t + s]; __syncthreads(); }
    if (t < 16) rs[t] = red[t];
}

// ----------------------------------------------- AP = A * P  (WMMA f32 16x16x4)
// 8 waves/block = 128 rows of A; the shared 2 KB P-chunk is staged into LDS
// once per block with double-buffered GLOBAL_LOAD_ASYNC_TO_LDS_B128, so the
// async DMA of chunk c+1 overlaps the WMMAs on chunk c. A rows stream straight
// from global (L2-resident, 159 MB) as b64 per-lane fragments.
__global__ void __launch_bounds__(256)
k_matvec(const float* __restrict__ A, const float* __restrict__ P,
         float* __restrict__ AP) {
    __shared__ __align__(16) float bbuf[2][KC * NPAD];   // 2 x 2 KB

    const int tid  = threadIdx.x;
    const int wave = tid >> 5, lane = tid & 31;
    const int tile = blockIdx.x * 8 + wave;
    const bool active = (tile < NDIM / 16);
    const int tilec = active ? tile : (NDIM / 16 - 1);   // clamp: stay in barriers
    const int r0 = tilec * 16;
    const int half = lane >> 4;          // A lanes 16-31 hold K=2,3
    const int m    = lane & 15;          // A row within tile / B column lane
    const float* arow = A + (long)(r0 + m) * NDIM + 2 * half;

    // LDS byte offsets of the two staging buffers (low 32 bits of generic ptr)
    const unsigned bb0 = (unsigned)(uintptr_t)&bbuf[0][0];
    const unsigned bb1 = (unsigned)(uintptr_t)&bbuf[1][0];

    // threads 0..127 each DMA one 16-byte quad of the 2 KB P-chunk
    const bool issuer = (tid < 128);
    if (issuer) {
        const float* g0 = P + 0 * (KC * NPAD) + tid * 4;
        unsigned l0 = bb0 + (unsigned)tid * 16u;
        asm volatile("global_load_async_to_lds_b128 %0, %1, off"
                     :: "v"(l0), "v"(g0) : "memory");
    }

    v8f acc = {};
    for (int c = 0; c < NCHUNK; ++c) {
        const int k0 = c * KC;
        const unsigned bbc = (c & 1) ? bb1 : bb0;
        if (issuer) {
            if (c + 1 < NCHUNK) {
                const float* gn = P + (size_t)(c + 1) * (KC * NPAD) + tid * 4;
                unsigned ln = ((c + 1) & 1 ? bb1 : bb0) + (unsigned)tid * 16u;
                asm volatile("global_load_async_to_lds_b128 %0, %1, off"
                             :: "v"(ln), "v"(gn) : "memory");
                asm volatile("s_wait_asynccnt 0x1" ::: "memory"); // chunk c landed
            } else {
                asm volatile("s_wait_asynccnt 0x0" ::: "memory");
            }
        }
        __syncthreads();

        const float* bc = (const float*)(bbuf[c & 1]);   // LDS view of chunk c
        __builtin_prefetch(arow + k0 + 1024, 0, 0);      // global_prefetch_b8
        #pragma unroll
        for (int kk = 0; kk < KC; kk += 4) {
            v2f a = *(const v2f*)(arow + k0 + kk);       // A[m][k+2h .. +1]
            v2f bfrag;
            bfrag.x = bc[(kk + half) * NPAD + m];        // B rows K=0/1 (ds_load)
            bfrag.y = bc[(kk + 2 + half) * NPAD + m];    // B rows K=2/3
            acc = __builtin_amdgcn_wmma_f32_16x16x4_f32(
                      /*neg_a=*/false, a, /*neg_b=*/false, bfrag,
                      /*c_mod=*/(short)0, acc, /*reuse_a=*/false, /*reuse_b=*/false);
        }
        __syncthreads();   // chunk c fully consumed before its buffer is re-DMA'd
    }

    if (active) {
        // C/D layout: VGPR j holds M=j (lanes 0-15) / M=j+8 (lanes 16-31), N=m
        #pragma unroll
        for (int j = 0; j < 8; ++j)
            AP[(long)(r0 + j + 8 * half) * NPAD + m] = acc[j];
    }
}

// ------------------------------------------------------- CG scalar/vector step
__global__ void __launch_bounds__(1024)
k_cgupd(float* __restrict__ X, float* __restrict__ R, float* __restrict__ Pv,
        float* __restrict__ AP, float* __restrict__ rs) {
    __shared__ float red[1024];
    __shared__ float s_alpha[16], s_beta[16];
    int t = threadIdx.x, col = t & 15, g = t >> 4;

    // pAp per batch column
    float part = 0.0f;
    for (int r = g; r < NDIM; r += 64) { int e = r * NPAD + col; part += Pv[e] * AP[e]; }
    red[t] = part; __syncthreads();
    for (int s = 512; s >= 16; s >>= 1) { if (t < s) red[t] += red[t + s]; __syncthreads(); }
    if (t < 16) {
        float pAp = red[t], r0 = rs[t];
        s_alpha[t] = (pAp != 0.0f) ? r0 / pAp : 0.0f;   // pad columns stay 0
    }
    __syncthreads();

    float alpha = s_alpha[col];
    float part2 = 0.0f;
    for (int r = g; r < NDIM; r += 64) {
        int e = r * NPAD + col;
        X[e] += alpha * Pv[e];
        float rn = R[e] - alpha * AP[e];
        R[e] = rn;
        part2 += rn * rn;
    }
    __syncthreads();
    red[t] = part2; __syncthreads();
    for (int s = 512; s >= 16; s >>= 1) { if (t < s) red[t] += red[t + s]; __syncthreads(); }
    if (t < 16) {
        float rn = red[t], r0 = rs[t];
        s_beta[t] = (r0 != 0.0f) ? rn / r0 : 0.0f;
        rs[t] = rn;
    }
    __syncthreads();
    float beta = s_beta[col];
    for (int r = g; r < NDIM; r += 64) { int e = r * NPAD + col; Pv[e] = R[e] + beta * Pv[e]; }
}

// -------------------------------------------------------------- classifier head
__global__ void __launch_bounds__(256)
k_head(const float* __restrict__ X, const float* __restrict__ wf,
       const float* __restrict__ bf, float* __restrict__ out) {
    int b = threadIdx.x >> 5, lane = threadIdx.x & 31;
    if (b >= BATCH) return;
    float logit = -1e30f;
    if (lane < 10) {
        logit = bf[lane];
        #pragma unroll
        for (int d = 0; d < DDIM; ++d)
            logit += wf[lane * DDIM + d] * X[d * NPAD + b];  // z[:,0] row = spin 0
    }
    float mx = -1e30f;
    for (int c = 0; c < 10; ++c) mx = fmaxf(mx, __shfl(logit, c, 32));
    float se = 0.0f;
    for (int c = 0; c < 10; ++c) se += __expf(__shfl(logit, c, 32) - mx);
    if (lane < 10) out[b * 10 + lane] = logit - mx - __logf(se);
}

// ------------------------------------------------------------------- driver
extern "C" void kernel_launch(void* const* d_in, const int* in_sizes, int n_in,
                              void* d_out, int out_size, void* d_ws, size_t ws_size,
                              hipStream_t stream) {
    const float* x   = (const float*)d_in[0];
    const float* wp  = (const float*)d_in[1];
    const float* bp  = (const float*)d_in[2];
    const float* w1  = (const float*)d_in[3];
    const float* b1  = (const float*)d_in[4];
    const float* w2  = (const float*)d_in[5];
    const float* b2  = (const float*)d_in[6];
    const float* cls = (const float*)d_in[7];
    const float* Jr  = (const float*)d_in[8];
    const float* wf  = (const float*)d_in[9];
    const float* bfin= (const float*)d_in[10];

    float* wsf = (float*)d_ws;
    float* A  = wsf;                               // 6304*6304 floats (~159 MB)
    float* T  = A  + (size_t)NDIM * NDIM;          // padded RHS  [NDIM][16]
    float* X  = T  + (size_t)NDIM * NPAD;
    float* R  = X  + (size_t)NDIM * NPAD;
    float* Pv = R  + (size_t)NDIM * NPAD;
    float* AP = Pv + (size_t)NDIM * NPAD;
    float* rs = AP + (size_t)NDIM * NPAD;          // 16 floats

    k_zero  <<<1024, 256, 0, stream>>>(T, 5L * NDIM * NPAD + 16);
    k_embed <<<NPATCH, 256, 0, stream>>>(x, wp, bp, w1, b1, w2, b2, T);
    k_buildA<<<8192, 256, 0, stream>>>(Jr, A);
    k_cginit<<<1, 1024, 0, stream>>>(T, X, R, Pv, rs, cls);

    const int tiles = NDIM / 16;                   // 394
    const int mvGrid = (tiles + 7) / 8;            // 8 waves / block
    for (int it = 0; it < CGIT; ++it) {
        k_matvec<<<mvGrid, 256, 0, stream>>>(A, Pv, AP);
        k_cgupd <<<1, 1024, 0, stream>>>(X, R, Pv, AP, rs);
    }
    k_head<<<1, 256, 0, stream>>>(X, wf, bfin, (float*)d_out);
}